// PartitionGNN_83451214561427
// MI455X (gfx1250) — compile-verified
//
#include <hip/hip_runtime.h>

// ---------------------------------------------------------------------------
// PartitionGNN for MI455X (gfx1250, wave32).
//   N = 100000 nodes, E = 1.6M edges, H = 64 features, G = 64 graphs.
// Edge phase: L2-resident f32 atomics (unsafeAtomicAdd -> global_atomic_add_f32,
//             no-return). Dense phase: V_WMMA_F32_16X16X4_F32 (exact f32 math).
// ---------------------------------------------------------------------------

typedef __attribute__((ext_vector_type(2))) float v2f;
typedef __attribute__((ext_vector_type(8))) float v8f;

#define HDIM 64

// ---- zero fill (float4 stores) --------------------------------------------
__global__ void gnn_zero_f4(float4* __restrict__ p, long long n4) {
  long long i = (long long)blockIdx.x * blockDim.x + threadIdx.x;
  if (i < n4) p[i] = make_float4(0.f, 0.f, 0.f, 0.f);
}

// ---- layer 0: scalar-feature scatter  agg0[dst] += ew * x[src] ------------
__global__ void gnn_scatter0(const int* __restrict__ src, const int* __restrict__ dst,
                             const float* __restrict__ ew, const float* __restrict__ x,
                             float* __restrict__ agg0, int E) {
  int e = blockIdx.x * blockDim.x + threadIdx.x;
  if (e < E) {
    float v = ew[e] * x[src[e]];
    unsafeAtomicAdd(&agg0[dst[e]], v);
  }
}

// ---- layer 0: rank-1 expand  h[i][h] = relu(agg0[i]*wrel[h] + b[h] + x[i]*wroot[h])
__global__ void gnn_expand0(const float* __restrict__ agg0, const float* __restrict__ x,
                            const float* __restrict__ wrel, const float* __restrict__ wroot,
                            const float* __restrict__ bias, float* __restrict__ hout,
                            int N) {
  long long idx = (long long)blockIdx.x * blockDim.x + threadIdx.x;
  int i = (int)(idx >> 6);
  int h = (int)(idx & 63);
  if (i < N) {
    float v = fmaf(agg0[i], wrel[h], fmaf(x[i], wroot[h], bias[h]));
    hout[idx] = fmaxf(v, 0.f);
  }
}

// ---- edge scatter, 64 features: 64 consecutive lanes share one edge -------
// -> coalesced no-return f32 atomics into the 25.6MB L2-resident agg buffer.
__global__ void gnn_scatter(const int* __restrict__ src, const int* __restrict__ dst,
                            const float* __restrict__ ew, const float* __restrict__ hin,
                            float* __restrict__ agg, long long total) {
  long long idx = (long long)blockIdx.x * blockDim.x + threadIdx.x;
  if (idx < total) {
    int e = (int)(idx >> 6);
    int h = (int)(idx & 63);
    float v = ew[e] * hin[((long long)src[e] << 6) + h];
    unsafeAtomicAdd(&agg[((long long)dst[e] << 6) + h], v);
  }
}

// ---- WMMA GEMM:  agg <- relu(agg @ wrel^T + hin @ wroot^T + b)  (in place)
// Block = 256 threads = 8 waves. Each wave owns a 16x16 tile of a 32x64
// output block. f32 WMMA 16x16x4, K = 64 -> 16 chained steps per term.
// Requires N % 32 == 0 (true: N = 100000) so EXEC is all ones for WMMA.
__global__ void __launch_bounds__(256)
gnn_gemm_root_relu(float* __restrict__ agg,        // [N][64] in: agg, out: h
                   const float* __restrict__ hin,  // [N][64] root input
                   const float* __restrict__ wrel, // [64][64] row-major W[n][k]
                   const float* __restrict__ wroot,// [64][64]
                   const float* __restrict__ bias) // [64]
{
  const int wave = threadIdx.x >> 5;   // 0..7
  const int lane = threadIdx.x & 31;
  const int half = lane >> 4;          // 0 | 1
  const int li   = lane & 15;
  const int rg   = wave >> 2;          // row group 0..1
  const int nt   = wave & 3;           // 16-col tile 0..3
  const int row0 = blockIdx.x * 32 + rg * 16;

  // A-fragments for the agg term: A[m=li][k = kb*4 + 2*half + v], v in {0,1}
  const float* aggRow = agg + ((long long)(row0 + li) << 6) + (half << 1);
  v2f aAgg[16];
#pragma unroll
  for (int kb = 0; kb < 16; ++kb)
    aAgg[kb] = *(const v2f*)(aggRow + kb * 4);

  // All in-place reads of this block's agg rows complete before any write.
  __syncthreads();

  const float* wrelRow  = wrel  + ((nt * 16 + li) << 6) + (half << 1);
  const float* wrootRow = wroot + ((nt * 16 + li) << 6) + (half << 1);
  const float* hinRow   = hin + ((long long)(row0 + li) << 6) + (half << 1);

  v8f c = {};
#pragma unroll
  for (int kb = 0; kb < 16; ++kb) {
    v2f b = *(const v2f*)(wrelRow + kb * 4);   // B[k][n] = wrel[n][k]
    c = __builtin_amdgcn_wmma_f32_16x16x4_f32(false, aAgg[kb], false, b,
                                              (short)0, c, false, false);
  }
#pragma unroll
  for (int kb = 0; kb < 16; ++kb) {
    v2f a = *(const v2f*)(hinRow + kb * 4);
    v2f b = *(const v2f*)(wrootRow + kb * 4);
    c = __builtin_amdgcn_wmma_f32_16x16x4_f32(false, a, false, b,
                                              (short)0, c, false, false);
  }

  // C/D layout: VGPR r -> row M = r + 8*half, col N = li.  Bias is per-N.
  const float bv = bias[nt * 16 + li];
  float* outp = agg + ((long long)row0 << 6) + nt * 16 + li;
#pragma unroll
  for (int r = 0; r < 8; ++r) {
    float v = fmaxf(c[r] + bv, 0.f);
    outp[(long long)(r + (half << 3)) << 6] = v;
  }
}

// ---- segment-mean pool accumulation ---------------------------------------
__global__ void gnn_pool(const float* __restrict__ h3, const int* __restrict__ batch,
                         float* __restrict__ sums, float* __restrict__ cnt, int N) {
  long long idx = (long long)blockIdx.x * blockDim.x + threadIdx.x;
  int i = (int)(idx >> 6);
  int h = (int)(idx & 63);
  if (i < N) {
    int g = batch[i];
    unsafeAtomicAdd(&sums[(g << 6) + h], h3[idx]);
    if (h == 0) unsafeAtomicAdd(&cnt[g], 1.0f);
  }
}

// ---- final linear: out[g] = dot(sums[g]/max(cnt,1), w_lin) + b_lin --------
__global__ void gnn_final(const float* __restrict__ sums, const float* __restrict__ cnt,
                          const float* __restrict__ wlin, const float* __restrict__ blin,
                          float* __restrict__ out, int G) {
  int g = threadIdx.x;
  if (g < G) {
    float c = fmaxf(cnt[g], 1.0f);
    float inv = 1.0f / c;
    float acc = blin[0];
#pragma unroll
    for (int h = 0; h < HDIM; ++h)
      acc = fmaf(sums[(g << 6) + h] * inv, wlin[h], acc);
    out[g] = acc;
  }
}

extern "C" void kernel_launch(void* const* d_in, const int* in_sizes, int n_in,
                              void* d_out, int out_size, void* d_ws, size_t ws_size,
                              hipStream_t stream) {
  const float* x   = (const float*)d_in[0];          // [N,1]
  const int* eidx  = (const int*)d_in[1];            // [2,E]
  const float* ew  = (const float*)d_in[2];          // [E,1]
  const int* batch = (const int*)d_in[3];            // [N]
  const float* wrel0  = (const float*)d_in[4];       // [64,1]
  const float* wroot0 = (const float*)d_in[5];       // [64,1]
  const float* b0     = (const float*)d_in[6];       // [64]
  const float* wrel1  = (const float*)d_in[7];       // [64,64]
  const float* wroot1 = (const float*)d_in[8];
  const float* b1     = (const float*)d_in[9];
  const float* wrel2  = (const float*)d_in[10];
  const float* wroot2 = (const float*)d_in[11];
  const float* b2     = (const float*)d_in[12];
  const float* wlin   = (const float*)d_in[13];      // [1,64]
  const float* blin   = (const float*)d_in[14];      // [1]
  float* out = (float*)d_out;

  const int N = in_sizes[0];             // 100000 (divisible by 32)
  const int E = in_sizes[1] / 2;         // 1600000
  const int G = out_size;                // 64
  const int* src = eidx;
  const int* dst = eidx + E;

  // workspace layout (all 16B aligned)
  char* ws = (char*)d_ws;
  const size_t nodeBytes = (size_t)N * HDIM * sizeof(float);   // 25.6 MB
  float* bufA = (float*)ws;                                    // h ping
  float* bufB = (float*)(ws + nodeBytes);                      // h pong
  float* agg0 = (float*)(ws + 2 * nodeBytes);                  // [N]
  float* sums = (float*)(ws + 2 * nodeBytes + (size_t)N * sizeof(float));
  float* cnt  = sums + (size_t)G * HDIM;                       // [G]

  const int T = 256;
  const long long NH  = (long long)N * HDIM;
  const long long EH  = (long long)E * HDIM;
  const long long gNH = (NH + T - 1) / T;
  const long long gEH = (EH + T - 1) / T;

  // ---- layer 0 (scalar features) ----
  gnn_zero_f4<<<(int)((N / 4 + T - 1) / T), T, 0, stream>>>((float4*)agg0, N / 4);
  gnn_scatter0<<<(E + T - 1) / T, T, 0, stream>>>(src, dst, ew, x, agg0, E);
  gnn_expand0<<<(int)gNH, T, 0, stream>>>(agg0, x, wrel0, wroot0, b0, bufA, N);

  // ---- layer 1: bufA -> bufB ----
  gnn_zero_f4<<<(int)((NH / 4 + T - 1) / T), T, 0, stream>>>((float4*)bufB, NH / 4);
  gnn_scatter<<<(int)gEH, T, 0, stream>>>(src, dst, ew, bufA, bufB, EH);
  gnn_gemm_root_relu<<<N / 32, T, 0, stream>>>(bufB, bufA, wrel1, wroot1, b1);

  // ---- layer 2: bufB -> bufA ----
  gnn_zero_f4<<<(int)((NH / 4 + T - 1) / T), T, 0, stream>>>((float4*)bufA, NH / 4);
  gnn_scatter<<<(int)gEH, T, 0, stream>>>(src, dst, ew, bufB, bufA, EH);
  gnn_gemm_root_relu<<<N / 32, T, 0, stream>>>(bufA, bufB, wrel2, wroot2, b2);

  // ---- pool + final linear ----
  const long long poolFloats = (long long)G * HDIM + G;        // sums + cnt
  gnn_zero_f4<<<(int)((poolFloats / 4 + T - 1) / T), T, 0, stream>>>(
      (float4*)sums, poolFloats / 4);
  gnn_pool<<<(int)gNH, T, 0, stream>>>(bufA, batch, sums, cnt, N);
  gnn_final<<<1, 64, 0, stream>>>(sums, cnt, wlin, blin, out, G);
}